// Drug_PNAConv_84318797955330
// MI455X (gfx1250) — compile-verified
//
#include <hip/hip_runtime.h>
#include <math.h>

// ---------------- problem constants ----------------
#define N_NODES 20000
#define N_EDGES 320000
#define HD      128          // hidden H
#define NT      4            // towers
#define FIN     32           // per-tower features
#define DEGBINS 128

typedef float v2f __attribute__((ext_vector_type(2)));
typedef float v8f __attribute__((ext_vector_type(8)));

// ---------------- workspace layout (float indices) ----------------
#define WS_AVG   0                          // 1 float: avg_deg log
#define WS_ENC   16                         // 5*FIN precomputed edge encodings
#define WS_CNT   256                        // N floats: in-degree count
#define WS_SUM   20480                      // N*H: segment sum
#define WS_SUM2  (WS_SUM  + N_NODES*HD)     // N*H: segment sum of squares
#define WS_MINB  (WS_SUM2 + N_NODES*HD)     // N*H uint: ordered-encoded min
#define WS_MAXB  (WS_MINB + N_NODES*HD)     // N*H uint: ordered-encoded max
#define WS_OUT2  (WS_MAXB + N_NODES*HD)     // N*H: post-tower-MLP output
// pair-packed weight copies (single-b64 B fragments)
#define WS_PW1P  (WS_OUT2 + N_NODES*HD)     // T*96*32  = 12288
#define WS_PW2P  (WS_PW1P + 12288)          // T*32*32  = 4096
#define WS_QW1P  (WS_PW2P + 4096)           // T*512*32 = 65536
#define WS_QW2P  (WS_QW1P + 65536)          // T*32*32  = 4096
#define WS_LINP  (WS_QW2P + 4096)           // 128*128  = 16384
// total ~12.92M floats (~51.7 MB)

// ---------------- WMMA helpers (V_WMMA_F32_16X16X4_F32) ----------------
__device__ __forceinline__ v8f wmma4(v2f a, v2f b, v8f c) {
  return __builtin_amdgcn_wmma_f32_16x16x4_f32(false, a, false, b, (short)0, c,
                                               false, false);
}

// A fragment from a row-major [16 x lda] tile (LDS): 16x4 f32 A layout.
__device__ __forceinline__ v2f frag_a(const float* A, int lda, int k0, int lane) {
  int row  = lane & 15;
  int koff = (lane >> 4) << 1;            // 0 or 2
  return *reinterpret_cast<const v2f*>(A + row * lda + k0 + koff);
}

// B fragment from a pair-packed weight matrix Bp[k/2][col][2]: one b64 load
__device__ __forceinline__ v2f frag_bp(const float* Bp, int ncols, int k0,
                                       int n0, int lane) {
  int col = n0 + (lane & 15);
  int k   = k0 + ((lane >> 4) << 1);
  return *reinterpret_cast<const v2f*>(Bp + (size_t)(k >> 1) * ncols * 2 + col * 2);
}

// order-preserving float <-> uint encoding for integer atomic min/max
__device__ __forceinline__ unsigned enc_ord(float f) {
  unsigned u = __float_as_uint(f);
  return (u & 0x80000000u) ? ~u : (u | 0x80000000u);
}
__device__ __forceinline__ float dec_ord(unsigned u) {
  return (u & 0x80000000u) ? __uint_as_float(u & 0x7FFFFFFFu)
                           : __uint_as_float(~u);
}

// async gather: one 16B chunk global -> LDS, tracked by ASYNCcnt
__device__ __forceinline__ void async_copy_b128(unsigned lds_off, const void* gsrc) {
  asm volatile("global_load_async_to_lds_b128 %0, %1, off"
               :: "v"(lds_off), "v"(gsrc) : "memory");
}
__device__ __forceinline__ void async_fence() {
  asm volatile("s_wait_asynccnt 0x0" ::: "memory");
}

// ---------------- kernel 0: init aggregation buffers ----------------
__global__ void pna_init(float* ws) {
  int i = blockIdx.x * blockDim.x + threadIdx.x;
  if (i < N_NODES * HD) {
    ws[WS_SUM  + i] = 0.f;
    ws[WS_SUM2 + i] = 0.f;
    reinterpret_cast<unsigned*>(ws)[WS_MINB + i] = 0xFFFFFFFFu;  // +inf (encoded)
    reinterpret_cast<unsigned*>(ws)[WS_MAXB + i] = 0u;           // -inf (encoded)
  }
  if (i < N_NODES) ws[WS_CNT + i] = 0.f;
}

// ---------------- pack: B[2p][c],B[2p+1][c] -> Bp[p][c][2] ----------------
__global__ void pna_pack(const float* __restrict__ src, float* __restrict__ dst,
                         int kpairs, int ncols) {
  int i = blockIdx.x * blockDim.x + threadIdx.x;
  if (i >= kpairs * ncols) return;
  int p = i / ncols, c = i % ncols;
  dst[(size_t)p * ncols * 2 + c * 2]     = src[(size_t)(2 * p)     * ncols + c];
  dst[(size_t)p * ncols * 2 + c * 2 + 1] = src[(size_t)(2 * p + 1) * ncols + c];
}

// ---------------- kernel 1: avg_log + enc_out[5][FIN] ----------------
__global__ void pna_prep(const int* __restrict__ mol_deg,
                         const float* __restrict__ bond_emb,
                         const float* __restrict__ enc_w,
                         const float* __restrict__ enc_b,
                         float* ws) {
  __shared__ float rn[256], rd[256];
  int t = threadIdx.x;
  float num = 0.f, den = 0.f;
  if (t < DEGBINS) {
    float d = (float)mol_deg[t];
    num = logf((float)t + 1.f) * d;
    den = d;
  }
  rn[t] = num; rd[t] = den;
  __syncthreads();
  for (int s = 128; s > 0; s >>= 1) {
    if (t < s) { rn[t] += rn[t + s]; rd[t] += rd[t + s]; }
    __syncthreads();
  }
  if (t == 0) ws[WS_AVG] = rn[0] / rd[0];
  for (int o = t; o < 5 * FIN; o += 256) {
    int i = o / FIN, j = o % FIN;
    float acc = enc_b[j];
    for (int k = 0; k < HD; ++k) acc += bond_emb[i * HD + k] * enc_w[k * FIN + j];
    ws[WS_ENC + o] = acc;
  }
}

// ---------------- kernel 2: per-edge pre-MLP (WMMA) + scatter aggregation ----
// block = 128 threads (4 waves); wave w handles tower w; tile = 16 edges
__global__ void pna_edge(const float* __restrict__ atom_x,
                         const int* __restrict__ bond_x,
                         const int* __restrict__ edge_index,
                         const float* __restrict__ pre_b1,
                         const float* __restrict__ pre_b2,
                         float* ws) {
  __shared__ float hT[NT][16][100];  // [tower][edge][3*FIN], pad keeps 16B align
  __shared__ float mt[NT][16][34];   // relu(pre1) staging
  __shared__ int   dstI[16];

  const int e0  = blockIdx.x * 16;
  const int tid = threadIdx.x;
  if (tid < 16) dstI[tid] = edge_index[N_EDGES + e0 + tid];

  // async gather of h = [x[dst] | x[src] | ea] straight into LDS, 16B chunks
  // NT*16*24 = 1536 chunks, 12 per thread, no divergence
  for (int idx = tid; idx < NT * 16 * 24; idx += 128) {
    int t   = idx / (16 * 24);
    int rem = idx % (16 * 24);
    int r = rem / 24, c = rem % 24;
    int e = e0 + r;
    const float* gsrc;
    if (c < 8)
      gsrc = atom_x + (size_t)edge_index[N_EDGES + e] * HD + t * FIN + c * 4;
    else if (c < 16)
      gsrc = atom_x + (size_t)edge_index[e] * HD + t * FIN + (c - 8) * 4;
    else
      gsrc = ws + WS_ENC + bond_x[e] * FIN + (c - 16) * 4;
    async_copy_b128((unsigned)(uintptr_t)&hT[t][r][c * 4], gsrc);
  }
  async_fence();
  __syncthreads();

  const int w    = tid >> 5;   // tower
  const int lane = tid & 31;
  const float* W1 = ws + WS_PW1P + (size_t)w * 96 * FIN;

  // GEMM1: [16 x 96] @ [96 x 32]
  v8f acc0 = {}, acc1 = {};
  for (int k0 = 0; k0 < 96; k0 += 4) {
    v2f a = frag_a(&hT[w][0][0], 100, k0, lane);
    acc0 = wmma4(a, frag_bp(W1, FIN, k0, 0,  lane), acc0);
    acc1 = wmma4(a, frag_bp(W1, FIN, k0, 16, lane), acc1);
  }
  const int col   = lane & 15;
  const int rbase = (lane < 16) ? 0 : 8;
  {
    float b1a = pre_b1[w * FIN + col];
    float b1b = pre_b1[w * FIN + 16 + col];
#pragma unroll
    for (int v = 0; v < 8; ++v) {
      mt[w][rbase + v][col]      = fmaxf(acc0[v] + b1a, 0.f);
      mt[w][rbase + v][16 + col] = fmaxf(acc1[v] + b1b, 0.f);
    }
  }
  __syncthreads();

  // GEMM2: [16 x 32] @ [32 x 32]
  const float* W2 = ws + WS_PW2P + (size_t)w * FIN * FIN;
  v8f z = {};
  acc0 = z; acc1 = z;
  for (int k0 = 0; k0 < FIN; k0 += 4) {
    v2f a = frag_a(&mt[w][0][0], 34, k0, lane);
    acc0 = wmma4(a, frag_bp(W2, FIN, k0, 0,  lane), acc0);
    acc1 = wmma4(a, frag_bp(W2, FIN, k0, 16, lane), acc1);
  }
  float b2a = pre_b2[w * FIN + col];
  float b2b = pre_b2[w * FIN + 16 + col];

  float*    sum  = ws + WS_SUM;
  float*    sum2 = ws + WS_SUM2;
  unsigned* minb = reinterpret_cast<unsigned*>(ws) + WS_MINB;
  unsigned* maxb = reinterpret_cast<unsigned*>(ws) + WS_MAXB;
#pragma unroll
  for (int v = 0; v < 8; ++v) {
    int r = rbase + v;
    size_t base = (size_t)dstI[r] * HD + w * FIN;
    float m0 = acc0[v] + b2a;
    float m1 = acc1[v] + b2b;
    atomicAdd(&sum [base + col],      m0);
    atomicAdd(&sum2[base + col],      m0 * m0);
    atomicMin(&minb[base + col],      enc_ord(m0));
    atomicMax(&maxb[base + col],      enc_ord(m0));
    atomicAdd(&sum [base + 16 + col], m1);
    atomicAdd(&sum2[base + 16 + col], m1 * m1);
    atomicMin(&minb[base + 16 + col], enc_ord(m1));
    atomicMax(&maxb[base + 16 + col], enc_ord(m1));
  }
  if (tid < 16) atomicAdd(ws + WS_CNT + dstI[tid], 1.0f);
}

// ---------------- kernel 3: per-node post-MLP (WMMA) ----------------
// grid (N/16, T); block = 32 threads (1 wave); tile = 16 nodes, one tower
__global__ void pna_post(const float* __restrict__ atom_x,
                         const float* __restrict__ post_b1,
                         const float* __restrict__ post_b2,
                         float* ws) {
  __shared__ float feat[16][514];   // [node][16*FIN], padded stride
  __shared__ float mt[16][34];

  const int t    = blockIdx.y;
  const int n0   = blockIdx.x * 16;
  const int lane = threadIdx.x;
  const float avg = ws[WS_AVG];

  // build the 512-wide per-tower feature: [x | agg | agg*s_amp | agg*s_att]
  for (int idx = lane; idx < 16 * 512; idx += 32) {
    int r = idx >> 9;
    int j = idx & 511;
    int n = n0 + r;
    float v;
    if (j < FIN) {
      v = atom_x[(size_t)n * HD + t * FIN + j];
    } else {
      int j2 = j - FIN;
      int g = j2 / 160;              // scaler group: id / amp / att
      int a = j2 % 160;
      int kind = a / FIN;            // sum/mean/min/max/std
      int c = a % FIN;
      size_t ib = (size_t)n * HD + t * FIN + c;
      float cnt  = ws[WS_CNT + n];
      float cnt1 = fmaxf(cnt, 1.f);
      float s    = ws[WS_SUM + ib];
      float base;
      if (kind == 0)      base = s;
      else if (kind == 1) base = s / cnt1;
      else if (kind == 2)
        base = (cnt > 0.f) ? dec_ord(reinterpret_cast<unsigned*>(ws)[WS_MINB + ib]) : 0.f;
      else if (kind == 3)
        base = (cnt > 0.f) ? dec_ord(reinterpret_cast<unsigned*>(ws)[WS_MAXB + ib]) : 0.f;
      else {
        float mean = s / cnt1;
        float m2   = ws[WS_SUM2 + ib] / cnt1;
        base = sqrtf(fmaxf(m2 - mean * mean, 0.f) + 1e-5f);
      }
      float sc = 1.f;
      if (g == 1)      sc = logf(cnt1 + 1.f) / avg;
      else if (g == 2) sc = avg / logf(cnt1 + 1.f);
      v = base * sc;
    }
    feat[r][j] = v;
  }
  __syncthreads();

  // GEMM1: [16 x 512] @ [512 x 32]
  const float* W1 = ws + WS_QW1P + (size_t)t * 512 * FIN;
  v8f acc0 = {}, acc1 = {};
  for (int k0 = 0; k0 < 512; k0 += 4) {
    v2f a = frag_a(&feat[0][0], 514, k0, lane);
    acc0 = wmma4(a, frag_bp(W1, FIN, k0, 0,  lane), acc0);
    acc1 = wmma4(a, frag_bp(W1, FIN, k0, 16, lane), acc1);
  }
  const int col   = lane & 15;
  const int rbase = (lane < 16) ? 0 : 8;
  {
    float b1a = post_b1[t * FIN + col];
    float b1b = post_b1[t * FIN + 16 + col];
#pragma unroll
    for (int v = 0; v < 8; ++v) {
      mt[rbase + v][col]      = fmaxf(acc0[v] + b1a, 0.f);
      mt[rbase + v][16 + col] = fmaxf(acc1[v] + b1b, 0.f);
    }
  }
  __syncthreads();

  // GEMM2: [16 x 32] @ [32 x 32]
  const float* W2 = ws + WS_QW2P + (size_t)t * FIN * FIN;
  v8f z = {};
  acc0 = z; acc1 = z;
  for (int k0 = 0; k0 < FIN; k0 += 4) {
    v2f a = frag_a(&mt[0][0], 34, k0, lane);
    acc0 = wmma4(a, frag_bp(W2, FIN, k0, 0,  lane), acc0);
    acc1 = wmma4(a, frag_bp(W2, FIN, k0, 16, lane), acc1);
  }
  float b2a = post_b2[t * FIN + col];
  float b2b = post_b2[t * FIN + 16 + col];
#pragma unroll
  for (int v = 0; v < 8; ++v) {
    int n = n0 + rbase + v;
    ws[WS_OUT2 + (size_t)n * HD + t * FIN + col]      = acc0[v] + b2a;
    ws[WS_OUT2 + (size_t)n * HD + t * FIN + 16 + col] = acc1[v] + b2b;
  }
}

// ---------------- kernel 4: final linear (WMMA) + LayerNorm + ReLU residual --
// grid N/16; block = 32 threads (1 wave); tile = 16 nodes
__global__ void pna_final(const float* __restrict__ atom_x,
                          const float* __restrict__ lin_b,
                          const float* __restrict__ ln_g,
                          const float* __restrict__ ln_b,
                          float* __restrict__ ws,
                          float* __restrict__ out) {
  __shared__ float ain[16][132];
  __shared__ float aout[16][132];

  const int n0   = blockIdx.x * 16;
  const int lane = threadIdx.x;

  // async copy of the 16x128 input tile straight into LDS (16B chunks)
  for (int idx = lane; idx < 16 * 32; idx += 32) {
    int r = idx >> 5, c = idx & 31;
    const float* gsrc = ws + WS_OUT2 + (size_t)(n0 + r) * HD + c * 4;
    async_copy_b128((unsigned)(uintptr_t)&ain[r][c * 4], gsrc);
  }
  async_fence();
  __syncthreads();

  const float* LW = ws + WS_LINP;
  const int col16 = lane & 15;
  const int rbase = (lane < 16) ? 0 : 8;
  for (int ct = 0; ct < 8; ++ct) {         // 8 column tiles of 16
    v8f acc = {};
    for (int k0 = 0; k0 < HD; k0 += 4) {
      v2f a = frag_a(&ain[0][0], 132, k0, lane);
      acc = wmma4(a, frag_bp(LW, HD, k0, ct * 16, lane), acc);
    }
    int col = ct * 16 + col16;
    float bb = lin_b[col];
#pragma unroll
    for (int v = 0; v < 8; ++v) aout[rbase + v][col] = acc[v] + bb;
  }
  __syncthreads();

  if (lane < 16) {
    int n = n0 + lane;
    float mu = 0.f;
    for (int c = 0; c < HD; ++c) mu += aout[lane][c];
    mu *= (1.f / HD);
    float var = 0.f;
    for (int c = 0; c < HD; ++c) { float d = aout[lane][c] - mu; var += d * d; }
    var *= (1.f / HD);
    float rinv = rsqrtf(var + 1e-5f);
    for (int c = 0; c < HD; ++c) {
      float ln = (aout[lane][c] - mu) * rinv * ln_g[c] + ln_b[c];
      out[(size_t)n * HD + c] = atom_x[(size_t)n * HD + c] + fmaxf(ln, 0.f);
    }
  }
}

// ---------------- host launcher ----------------
extern "C" void kernel_launch(void* const* d_in, const int* in_sizes, int n_in,
                              void* d_out, int out_size, void* d_ws, size_t ws_size,
                              hipStream_t stream) {
  const float* atom_x   = (const float*)d_in[0];
  const int*   bond_x   = (const int*)  d_in[1];
  const int*   edge_idx = (const int*)  d_in[2];
  const int*   mol_deg  = (const int*)  d_in[3];
  const float* bond_emb = (const float*)d_in[4];
  const float* enc_w    = (const float*)d_in[5];
  const float* enc_b    = (const float*)d_in[6];
  const float* pre_w1   = (const float*)d_in[7];
  const float* pre_b1   = (const float*)d_in[8];
  const float* pre_w2   = (const float*)d_in[9];
  const float* pre_b2   = (const float*)d_in[10];
  const float* post_w1  = (const float*)d_in[11];
  const float* post_b1  = (const float*)d_in[12];
  const float* post_w2  = (const float*)d_in[13];
  const float* post_b2  = (const float*)d_in[14];
  const float* lin_w    = (const float*)d_in[15];
  const float* lin_b    = (const float*)d_in[16];
  const float* ln_g     = (const float*)d_in[17];
  const float* ln_b     = (const float*)d_in[18];

  float* ws  = (float*)d_ws;
  float* out = (float*)d_out;

  pna_init<<<(N_NODES * HD + 255) / 256, 256, 0, stream>>>(ws);
  pna_prep<<<1, 256, 0, stream>>>(mol_deg, bond_emb, enc_w, enc_b, ws);

  // pair-pack all weight matrices (K even everywhere)
  pna_pack<<<(192  * 32  + 255) / 256, 256, 0, stream>>>(pre_w1,  ws + WS_PW1P, 192,  32);
  pna_pack<<<(64   * 32  + 255) / 256, 256, 0, stream>>>(pre_w2,  ws + WS_PW2P, 64,   32);
  pna_pack<<<(1024 * 32  + 255) / 256, 256, 0, stream>>>(post_w1, ws + WS_QW1P, 1024, 32);
  pna_pack<<<(64   * 32  + 255) / 256, 256, 0, stream>>>(post_w2, ws + WS_QW2P, 64,   32);
  pna_pack<<<(64   * 128 + 255) / 256, 256, 0, stream>>>(lin_w,   ws + WS_LINP, 64,   128);

  pna_edge<<<N_EDGES / 16, 128, 0, stream>>>(atom_x, bond_x, edge_idx,
                                             pre_b1, pre_b2, ws);
  pna_post<<<dim3(N_NODES / 16, NT), 32, 0, stream>>>(atom_x, post_b1, post_b2, ws);
  pna_final<<<N_NODES / 16, 32, 0, stream>>>(atom_x, lin_b, ln_g, ln_b, ws, out);
}